// LossWithEuler_27771258536669
// MI455X (gfx1250) — compile-verified
//
#include <hip/hip_runtime.h>
#include <math.h>

typedef __attribute__((ext_vector_type(4))) float f4;
typedef __attribute__((ext_vector_type(2))) float v2f;
typedef __attribute__((ext_vector_type(8))) float v8f;

#define D_COLS 512
#define ROWS_PER_BLOCK 16
#define THREADS 256
#define F4_PER_ROW (D_COLS / 4)                      // 128
#define F4_PER_TILE (ROWS_PER_BLOCK * F4_PER_ROW)    // 2048
#define ITERS (F4_PER_TILE / THREADS)                // 8

// One block handles a 16-row x 512-col tile. Weights for the 16 rows are
// computed into LDS by the first 16 threads (fuses the per-row weight pass).
// Main loop: coalesced non-temporal b128 loads, per-lane f32 accumulation.
// Wave reduction done via two V_WMMA_F32_16X16X4_F32 against an all-ones B.
__global__ __launch_bounds__(THREADS) void loss_main_kernel(
    const float* __restrict__ inp, const float* __restrict__ label,
    const float* __restrict__ ea, const int* __restrict__ attr,
    const float* __restrict__ attr_num, float* __restrict__ block_sums) {
  __shared__ float sw[ROWS_PER_BLOCK];
  __shared__ float wave_part[THREADS / 32];

  const int t = threadIdx.x;
  const int r0 = blockIdx.x * ROWS_PER_BLOCK;

  if (t < ROWS_PER_BLOCK) {
    const int r = r0 + t;
    // inverse-frequency weights: sum(attribute_num) / attribute_num[j]
    float an[6], total = 0.0f;
#pragma unroll
    for (int j = 0; j < 6; ++j) { an[j] = attr_num[j]; total += an[j]; }
    float attr_w = 0.0f;
#pragma unroll
    for (int j = 0; j < 6; ++j)
      attr_w += (attr[r * 6 + j] == 1) ? (total / an[j]) : 0.0f;
    float ang_w = (1.0f - cosf(ea[r * 3 + 0])) +
                  (1.0f - cosf(ea[r * 3 + 1])) +
                  (1.0f - cosf(ea[r * 3 + 2]));
    sw[t] = ang_w * attr_w;
  }
  __syncthreads();

  const f4* __restrict__ inp4 =
      reinterpret_cast<const f4*>(inp) + (size_t)r0 * F4_PER_ROW;
  const f4* __restrict__ lab4 =
      reinterpret_cast<const f4*>(label) + (size_t)r0 * F4_PER_ROW;

  f4 acc = {0.0f, 0.0f, 0.0f, 0.0f};
#pragma unroll
  for (int i = 0; i < ITERS; ++i) {
    const int idx = t + i * THREADS;          // consecutive threads -> consecutive float4s
    const int row_local = idx >> 7;           // idx / F4_PER_ROW
    f4 a = __builtin_nontemporal_load(&inp4[idx]);
    f4 b = __builtin_nontemporal_load(&lab4[idx]);
    f4 d = a - b;
    const float w = sw[row_local];
    f4 w4 = {w, w, w, w};
    acc += w4 * d * d;
  }

  // --- WMMA wave reduction -------------------------------------------------
  // 32 lanes x 4 partials = 128 values = two 16x4 f32 A tiles.
  // With B = all-ones, D[m,n] = sum_k A[m,k] (replicated over n).
  // Per-lane sum of 8 D-VGPRs covers one M-half; lane L^16 holds the other.
  v2f ones = {1.0f, 1.0f};
  v2f a0 = {acc.x, acc.y};
  v2f a1 = {acc.z, acc.w};
  v8f c = {};
  c = __builtin_amdgcn_wmma_f32_16x16x4_f32(false, a0, false, ones,
                                            (short)0, c, false, false);
  c = __builtin_amdgcn_wmma_f32_16x16x4_f32(false, a1, false, ones,
                                            (short)0, c, false, false);
  float half_sum = c[0] + c[1] + c[2] + c[3] + c[4] + c[5] + c[6] + c[7];
  float wave_sum = half_sum + __shfl_xor(half_sum, 16, 32);

  if ((t & 31) == 0) wave_part[t >> 5] = wave_sum;
  __syncthreads();
  if (t == 0) {
    float s = 0.0f;
#pragma unroll
    for (int wv = 0; wv < THREADS / 32; ++wv) s += wave_part[wv];
    block_sums[blockIdx.x] = s;
  }
}

// Deterministic fixed-order reduction of per-block sums -> final mean.
__global__ __launch_bounds__(256) void final_reduce_kernel(
    const float* __restrict__ block_sums, int n, float scale,
    float* __restrict__ out) {
  __shared__ float smem[256];
  float acc = 0.0f;
  for (int i = threadIdx.x; i < n; i += 256) acc += block_sums[i];
  smem[threadIdx.x] = acc;
  __syncthreads();
  for (int off = 128; off > 0; off >>= 1) {
    if ((int)threadIdx.x < off) smem[threadIdx.x] += smem[threadIdx.x + off];
    __syncthreads();
  }
  if (threadIdx.x == 0) out[0] = smem[0] * scale;
}

extern "C" void kernel_launch(void* const* d_in, const int* in_sizes, int n_in,
                              void* d_out, int out_size, void* d_ws,
                              size_t ws_size, hipStream_t stream) {
  const float* inp = (const float*)d_in[0];
  const float* label = (const float*)d_in[1];
  const float* ea = (const float*)d_in[2];
  const int* attr = (const int*)d_in[3];
  const float* attr_num = (const float*)d_in[4];
  (void)n_in; (void)out_size; (void)ws_size;

  const int B = in_sizes[0] / D_COLS;          // 32768
  const int grid = B / ROWS_PER_BLOCK;         // 2048 blocks

  float* block_sums = (float*)d_ws;            // grid floats of scratch

  loss_main_kernel<<<grid, THREADS, 0, stream>>>(inp, label, ea, attr,
                                                 attr_num, block_sums);

  const float scale = 1.0f / ((float)B * (float)D_COLS);
  final_reduce_kernel<<<1, 256, 0, stream>>>(block_sums, grid, scale,
                                             (float*)d_out);
}